// DominantNeigh_70068096467662
// MI455X (gfx1250) — compile-verified
//
#include <hip/hip_runtime.h>
#include <hip/hip_bf16.h>

#define N_NODES 16384
#define FEAT    512
#define HID     128
#define NEDGE   524288
#define BETA    0.001f

typedef float v2f __attribute__((ext_vector_type(2)));
typedef float v8f __attribute__((ext_vector_type(8)));
typedef __attribute__((__vector_size__(4 * sizeof(int)))) int i4v;

// gfx1250 async global->LDS path if the toolchain exposes it
#if defined(__has_builtin)
#  if __has_builtin(__builtin_amdgcn_global_load_async_to_lds_b128) && \
      __has_builtin(__builtin_amdgcn_s_wait_asynccnt)
#    define USE_ASYNC_LDS 1
#  endif
#endif
#ifndef USE_ASYNC_LDS
#  define USE_ASYNC_LDS 0
#endif

// D = A(16x4, f32) * B(4x16, f32) + C(16x16, f32)  -- CDNA5 V_WMMA_F32_16X16X4_F32
__device__ __forceinline__ v8f wmma_f32x4(v2f a, v2f b, v8f c) {
  return __builtin_amdgcn_wmma_f32_16x16x4_f32(
      /*neg_a=*/false, a, /*neg_b=*/false, b,
      /*c_mod=*/(short)0, c, /*reuse_a=*/false, /*reuse_b=*/false);
}

// ---------------------------------------------------------------------------
// Generic GEMM: C[M,Nc] = A[M,K] @ W[K,Nc] (+ bias[Nc] if bias != null)
// Row-major everything. One wave computes one 16x16 output tile.
// Fragment layouts per CDNA5 ISA 7.12.2:
//   A: lane l holds A[m = l&15][k = 2*(l>>4) + v], v in {0,1}
//   B: lane l holds W[k = 2*(l>>4) + v][n = l&15]
//   C: lane l holds C[m = v + 8*(l>>4)][n = l&15], v in 0..7
// NT=true streams the result past L2 (write-once outputs).
// ---------------------------------------------------------------------------
template <bool NT>
__global__ void gemm_wmma_bias(const float* __restrict__ A,
                               const float* __restrict__ W,
                               const float* __restrict__ bias,
                               float* __restrict__ C,
                               int M, int K, int Nc) {
  const int lane   = threadIdx.x & 31;
  const int wid    = (blockIdx.x * blockDim.x + threadIdx.x) >> 5;
  const int tilesN = Nc >> 4;
  const int tm     = wid / tilesN;
  const int tn     = wid - tm * tilesN;
  if (tm * 16 >= M) return;

  const int r = lane & 15;
  const int h = lane >> 4;               // half select (0/1)
  const int m = tm * 16 + r;
  const int n = tn * 16 + r;
  const float* arow = A + (size_t)m * K;

  v8f acc = {};
  for (int k0 = 0; k0 < K; k0 += 4) {
    const int ka = k0 + 2 * h;           // even -> 8B aligned
    v2f a = *(const v2f*)(arow + ka);
    v2f b;
    b.x = W[(size_t)(ka + 0) * Nc + n];
    b.y = W[(size_t)(ka + 1) * Nc + n];
    acc = wmma_f32x4(a, b, acc);
  }

  const float bn = bias ? bias[n] : 0.0f;
#pragma unroll
  for (int v = 0; v < 8; ++v) {
    const int row = tm * 16 + v + 8 * h;
    float* dst = C + (size_t)row * Nc + n;
    if (NT) __builtin_nontemporal_store(acc[v] + bn, dst);
    else    *dst = acc[v] + bn;
  }
}

// ---------------------------------------------------------------------------
// a_hat = S @ S^T, S is [N_NODES, HID] row-major.
// Block = 256 threads (8 waves) computes a 64x64 output tile.
// A-rows and B-rows (64 x 128 f32 each) are staged in LDS with a 132-float
// row pitch: 132 mod 64 = 4, so the 16 lanes of a fragment read hit 16
// distinct banks (lanes 16-31 hit the disjoint +2 bank set) -> conflict-free
// ds_load_b64. Each wave owns one 16-row M-subtile and two 16-col N-subtiles.
// Global traffic: 64KB load per 16KB store; output streamed with NT stores
// so the 1.07GB write-once a_hat doesn't evict the L2-resident S matrix.
// ---------------------------------------------------------------------------
#define BT 64
#define LPITCH 132
#define AAT_SMEM_BYTES (2 * BT * LPITCH * 4)

__global__ void gemm_aat_wmma(const float* __restrict__ S,
                              float* __restrict__ C) {
  extern __shared__ float smem[];
  float* As = smem;                  // [64][132]
  float* Bs = smem + BT * LPITCH;    // [64][132]

  const int tid = threadIdx.x;
  const int nbt = N_NODES / BT;      // 256 block tiles per dim
  const int bm  = blockIdx.x / nbt;
  const int bn  = blockIdx.x - bm * nbt;
  const int m0  = bm * BT;
  const int n0  = bn * BT;

  // Stage 64x128 A-rows and B-rows; thread t copies float4 chunks, coalesced.
#pragma unroll
  for (int it = 0; it < 8; ++it) {
    const int linear = it * 256 + tid;
    const int row = linear >> 5;
    const int col = (linear & 31) * 4;
    const float* gA = S + (size_t)(m0 + row) * HID + col;
    const float* gB = S + (size_t)(n0 + row) * HID + col;
    float* lA = As + row * LPITCH + col;
    float* lB = Bs + row * LPITCH + col;
#if USE_ASYNC_LDS
    __builtin_amdgcn_global_load_async_to_lds_b128((i4v*)gA, (i4v*)lA, 0, 0);
    __builtin_amdgcn_global_load_async_to_lds_b128((i4v*)gB, (i4v*)lB, 0, 0);
#else
    *(float4*)lA = *(const float4*)gA;
    *(float4*)lB = *(const float4*)gB;
#endif
  }
#if USE_ASYNC_LDS
  __builtin_amdgcn_s_wait_asynccnt(0);
#endif
  __syncthreads();

  const int lane = tid & 31;
  const int wav  = tid >> 5;          // 0..7
  const int r    = lane & 15;
  const int h    = lane >> 4;
  const int wm   = wav >> 1;          // 0..3  (M subtile)
  const int wn   = (wav & 1) * 2;     // 0 or 2 (base N subtile; wave does 2)

  const float* a_base  = As + (wm * 16 + r) * LPITCH;
  const float* b_base0 = Bs + ((wn + 0) * 16 + r) * LPITCH;
  const float* b_base1 = Bs + ((wn + 1) * 16 + r) * LPITCH;

  v8f acc0 = {}, acc1 = {};
#pragma unroll
  for (int k0 = 0; k0 < HID; k0 += 4) {
    const int ka = k0 + 2 * h;
    const v2f a  = *(const v2f*)(a_base + ka);
    const v2f b0 = *(const v2f*)(b_base0 + ka);
    const v2f b1 = *(const v2f*)(b_base1 + ka);
    acc0 = wmma_f32x4(a, b0, acc0);
    acc1 = wmma_f32x4(a, b1, acc1);
  }

#pragma unroll
  for (int v = 0; v < 8; ++v) {
    const size_t row = (size_t)(m0 + wm * 16 + v + 8 * h);
    float* crow = C + row * (size_t)N_NODES + n0;
    __builtin_nontemporal_store(acc0[v], crow + (wn + 0) * 16 + r);
    __builtin_nontemporal_store(acc1[v], crow + (wn + 1) * 16 + r);
  }
}

// ---------------------------------------------------------------------------
__global__ void zero_f4(float4* __restrict__ p, int n4) {
  const int i = blockIdx.x * blockDim.x + threadIdx.x;
  if (i < n4) p[i] = make_float4(0.f, 0.f, 0.f, 0.f);
}

// msgs = support[src] * w ; agg[dst] += msgs  (32 threads per edge, 4 floats each)
__global__ void edge_scatter(const float* __restrict__ support,
                             const int* __restrict__ ei,     // [2, E]: row0=dst, row1=src
                             const float* __restrict__ ew,
                             float* __restrict__ agg) {
  const unsigned t = blockIdx.x * blockDim.x + threadIdx.x;
  const unsigned e = t >> 5;
  if (e >= NEDGE) return;
  const unsigned c = (t & 31u) * 4u;
  const int dst = ei[e];
  const int src = ei[NEDGE + e];
  const float w = ew[e];
  const float4 v = *(const float4*)(support + (size_t)src * HID + c);
  float* o = agg + (size_t)dst * HID + c;
  atomicAdd(o + 0, v.x * w);
  atomicAdd(o + 1, v.y * w);
  atomicAdd(o + 2, v.z * w);
  atomicAdd(o + 3, v.w * w);
}

// out = relu( (beta*h + (1-beta)*agg) / max(||.||2, eps) + bias )
// one wave32 per row; lane handles 4 of the 128 features
__global__ void residual_norm_bias_relu(const float* __restrict__ hin,
                                        const float* __restrict__ agg,
                                        const float* __restrict__ bias,
                                        float* __restrict__ hout) {
  const int wpb  = blockDim.x >> 5;
  const int row  = blockIdx.x * wpb + (threadIdx.x >> 5);
  const int lane = threadIdx.x & 31;
  if (row >= N_NODES) return;

  const size_t base = (size_t)row * HID + lane * 4;
  const float4 hv = *(const float4*)(hin + base);
  const float4 av = *(const float4*)(agg + base);
  float4 o;
  o.x = BETA * hv.x + (1.0f - BETA) * av.x;
  o.y = BETA * hv.y + (1.0f - BETA) * av.y;
  o.z = BETA * hv.z + (1.0f - BETA) * av.z;
  o.w = BETA * hv.w + (1.0f - BETA) * av.w;

  float ss = o.x * o.x + o.y * o.y + o.z * o.z + o.w * o.w;
#pragma unroll
  for (int off = 16; off > 0; off >>= 1) ss += __shfl_xor(ss, off, 32);

  const float inv = 1.0f / fmaxf(sqrtf(ss), 1e-12f);
  const float4 b4 = *(const float4*)(bias + lane * 4);
  o.x = fmaxf(o.x * inv + b4.x, 0.0f);
  o.y = fmaxf(o.y * inv + b4.y, 0.0f);
  o.z = fmaxf(o.z * inv + b4.z, 0.0f);
  o.w = fmaxf(o.w * inv + b4.w, 0.0f);
  *(float4*)(hout + base) = o;
}

// ---------------------------------------------------------------------------
static void launch_gcn(const float* h_in, const float* W, const float* b,
                       const int* ei, const float* ew,
                       float* support, float* agg, float* h_out,
                       hipStream_t stream) {
  // support = h_in @ W   (16384x128 @ 128x128) -> 8192 tiles, 8 waves/block
  gemm_wmma_bias<false><<<1024, 256, 0, stream>>>(h_in, W, nullptr, support,
                                                  N_NODES, HID, HID);
  // agg = 0
  zero_f4<<<2048, 256, 0, stream>>>((float4*)agg, (N_NODES * HID) / 4);
  // agg[dst] += support[src] * w
  edge_scatter<<<65536, 256, 0, stream>>>(support, ei, ew, agg);
  // h_out = relu(normalize(beta*h_in + (1-beta)*agg) + b)
  residual_norm_bias_relu<<<2048, 256, 0, stream>>>(h_in, agg, b, h_out);
}

extern "C" void kernel_launch(void* const* d_in, const int* in_sizes, int n_in,
                              void* d_out, int out_size, void* d_ws, size_t ws_size,
                              hipStream_t stream) {
  const float* x          = (const float*)d_in[0];
  const int*   ei         = (const int*)  d_in[1];
  const float* ew         = (const float*)d_in[2];
  const float* enc_lin_W  = (const float*)d_in[3];
  const float* enc_lin_b  = (const float*)d_in[4];
  const float* enc_gc1_W  = (const float*)d_in[5];
  const float* enc_gc1_b  = (const float*)d_in[6];
  const float* enc_gc2_W  = (const float*)d_in[7];
  const float* enc_gc2_b  = (const float*)d_in[8];
  const float* attr_gc1_W = (const float*)d_in[9];
  const float* attr_gc1_b = (const float*)d_in[10];
  const float* attr_gc2_W = (const float*)d_in[11];
  const float* attr_gc2_b = (const float*)d_in[12];
  const float* attr_lin_W = (const float*)d_in[13];
  const float* attr_lin_b = (const float*)d_in[14];
  const float* str_gc1_W  = (const float*)d_in[15];
  const float* str_gc1_b  = (const float*)d_in[16];

  float* a_hat = (float*)d_out;                                  // [N, N]
  float* x_hat = (float*)d_out + (size_t)N_NODES * N_NODES;      // [N, FEAT]

  const size_t NH = (size_t)N_NODES * HID;
  float* ws      = (float*)d_ws;
  float* bufA    = ws;            // activation ping
  float* bufB    = ws + 1 * NH;   // activation pong
  float* bufC    = ws + 2 * NH;   // activation third
  float* support = ws + 3 * NH;
  float* agg     = ws + 4 * NH;   // total 5 * 8MB = 40MB

  // ---- Encoder ----
  // bufA = x @ enc_lin_W + b : 16384x512 @ 512x128 -> 8192 tiles
  gemm_wmma_bias<false><<<1024, 256, 0, stream>>>(x, enc_lin_W, enc_lin_b, bufA,
                                                  N_NODES, FEAT, HID);
  launch_gcn(bufA, enc_gc1_W, enc_gc1_b, ei, ew, support, agg, bufB, stream);
  launch_gcn(bufB, enc_gc2_W, enc_gc2_b, ei, ew, support, agg, bufA, stream);
  const float* h_enc = bufA;

  // ---- Attribute decoder ----
  launch_gcn(h_enc, attr_gc1_W, attr_gc1_b, ei, ew, support, agg, bufB, stream);
  launch_gcn(bufB,  attr_gc2_W, attr_gc2_b, ei, ew, support, agg, bufC, stream);
  // x_hat = bufC @ attr_lin_W + b : 16384x128 @ 128x512 -> 32768 tiles (NT out)
  gemm_wmma_bias<true><<<4096, 256, 0, stream>>>(bufC, attr_lin_W, attr_lin_b, x_hat,
                                                 N_NODES, HID, FEAT);

  // ---- Structure decoder ----
  launch_gcn(h_enc, str_gc1_W, str_gc1_b, ei, ew, support, agg, bufB, stream);
  // a_hat = s @ s^T : 256x256 block tiles of 64x64, 8 waves each
  gemm_aat_wmma<<<65536, 256, AAT_SMEM_BYTES, stream>>>(bufB, a_hat);
}